// SimpleGCN_52836687675913
// MI455X (gfx1250) — compile-verified
//
#include <hip/hip_runtime.h>

typedef __attribute__((ext_vector_type(2))) float v2f;
typedef __attribute__((ext_vector_type(4))) float v4f;
typedef __attribute__((ext_vector_type(8))) float v8f;

// ---------------------------------------------------------------------------
// C[M,N] = op(A[M,K]) @ B[K,N],  op = optional ReLU applied to A on load.
// f32 WMMA 16x16x4 (exact-f32 shape). blockDim = 128 (4 wave32s).
// Each wave computes a 32x32 register tile = 2x2 WMMA tiles -> 4 independent
// accumulator chains (hides WMMA latency), sharing A across the col pair and
// B across the row pair: 6 loads feed 4 WMMAs per K-step.
//   rows: blockIdx.x*32 .. +31 ; cols: (wave)*32 within blockIdx.y*128 strip.
// A frag (16x4 f32, 2 VGPR): lanes 0-15 -> K=k0,k0+1 ; lanes 16-31 -> k0+2,k0+3
// B frag (4x16 f32, 2 VGPR): same K split, N striped across lanes 0-15.
// C/D (16x16 f32, 8 VGPR): VGPR r -> row = row0 + (lane>>4)*8 + r, col = lane&15.
// ---------------------------------------------------------------------------
template <bool RELU_A>
__global__ __launch_bounds__(128) void gemm_wmma_f32(
    const float* __restrict__ A, const float* __restrict__ B,
    float* __restrict__ C, int M, int N, int K) {
  const int lane = threadIdx.x & 31;
  const int wave = threadIdx.x >> 5;
  const int row0 = blockIdx.x * 32;
  const int col0 = (blockIdx.y * 4 + wave) * 32;
  if (col0 >= N || row0 >= M) return;  // wave-uniform: EXEC stays all-1s

  const int mr0 = row0 + (lane & 15);
  const int mr1 = mr0 + 16;
  const int ar0 = (mr0 < M) ? mr0 : (M - 1);  // clamped safe reads
  const int ar1 = (mr1 < M) ? mr1 : (M - 1);
  const int koff = (lane >> 4) * 2;           // 0 or 2
  const int bc0 = col0 + (lane & 15);
  const int bc1 = bc0 + 16;

  const float* __restrict__ arow0 = A + (size_t)ar0 * K + koff;
  const float* __restrict__ arow1 = A + (size_t)ar1 * K + koff;

  v8f acc00 = {}, acc01 = {}, acc10 = {}, acc11 = {};
  for (int k0 = 0; k0 < K; k0 += 4) {
    v2f a0 = *(const v2f*)(arow0 + k0);
    v2f a1 = *(const v2f*)(arow1 + k0);
    if (RELU_A) {
      a0.x = fmaxf(a0.x, 0.0f); a0.y = fmaxf(a0.y, 0.0f);
      a1.x = fmaxf(a1.x, 0.0f); a1.y = fmaxf(a1.y, 0.0f);
    }
    const float* brow = B + (size_t)(k0 + koff) * N;
    v2f b0, b1;
    b0.x = brow[bc0]; b0.y = brow[N + bc0];
    b1.x = brow[bc1]; b1.y = brow[N + bc1];
    acc00 = __builtin_amdgcn_wmma_f32_16x16x4_f32(false, a0, false, b0, (short)0, acc00, false, false);
    acc01 = __builtin_amdgcn_wmma_f32_16x16x4_f32(false, a0, false, b1, (short)0, acc01, false, false);
    acc10 = __builtin_amdgcn_wmma_f32_16x16x4_f32(false, a1, false, b0, (short)0, acc10, false, false);
    acc11 = __builtin_amdgcn_wmma_f32_16x16x4_f32(false, a1, false, b1, (short)0, acc11, false, false);
  }

  const int rb0 = row0 + (lane >> 4) * 8;       // rows for acc0x
  const int rb1 = rb0 + 16;                     // rows for acc1x
#pragma unroll
  for (int r = 0; r < 8; ++r) {
    const int r0 = rb0 + r, r1 = rb1 + r;
    if (r0 < M) {
      C[(size_t)r0 * N + bc0] = acc00[r];
      C[(size_t)r0 * N + bc1] = acc01[r];
    }
    if (r1 < M) {
      C[(size_t)r1 * N + bc0] = acc10[r];
      C[(size_t)r1 * N + bc1] = acc11[r];
    }
  }
}

// ---------------------------------------------------------------------------
// agg[n][f] = bias[f]  -- vectorized b128 broadcast init
// ---------------------------------------------------------------------------
__global__ __launch_bounds__(256) void init_bias4(
    float* __restrict__ out, const float* __restrict__ bias, int total4, int f4mask) {
  const int i = blockIdx.x * blockDim.x + threadIdx.x;
  if (i < total4)
    ((v4f*)out)[i] = ((const v4f*)bias)[i & f4mask];
}

// ---------------------------------------------------------------------------
// For each edge e:  agg[dst[e]][:] += h[src[e]][:]
// One wave per edge; edge id forced uniform (readfirstlane) so src/dst become
// scalar SMEM loads. Each lane handles VPT = F/32 consecutive features via one
// vector load, then VPT f32 atomics to consecutive addresses (L2-coalesced;
// the 51 MB feature matrix is resident in the 192 MB L2).
// ---------------------------------------------------------------------------
template <int VPT>
__global__ __launch_bounds__(256) void scatter_add(
    float* __restrict__ agg, const float* __restrict__ h,
    const long long* __restrict__ src, const long long* __restrict__ dst,
    int E) {
  const int lane = threadIdx.x & 31;
  const int e = __builtin_amdgcn_readfirstlane(
      blockIdx.x * (blockDim.x >> 5) + (threadIdx.x >> 5));
  if (e >= E) return;
  const int s = __builtin_amdgcn_readfirstlane((int)src[e]);
  const int d = __builtin_amdgcn_readfirstlane((int)dst[e]);
  const int F = VPT * 32;
  const int f0 = lane * VPT;

  float v[VPT];
  *(float4*)nullptr;  // (never executed; placeholder removed below)
}

// Specialized loads to get clean b128/b64 vector codegen.
template <>
__global__ __launch_bounds__(256) void scatter_add<4>(
    float* __restrict__ agg, const float* __restrict__ h,
    const long long* __restrict__ src, const long long* __restrict__ dst,
    int E) {
  const int lane = threadIdx.x & 31;
  const int e = __builtin_amdgcn_readfirstlane(
      blockIdx.x * (blockDim.x >> 5) + (threadIdx.x >> 5));
  if (e >= E) return;
  const int s = __builtin_amdgcn_readfirstlane((int)src[e]);
  const int d = __builtin_amdgcn_readfirstlane((int)dst[e]);
  const int f0 = lane * 4;
  const v4f v = *(const v4f*)(h + (size_t)s * 128 + f0);
  float* p = agg + (size_t)d * 128 + f0;
  unsafeAtomicAdd(p + 0, v.x);
  unsafeAtomicAdd(p + 1, v.y);
  unsafeAtomicAdd(p + 2, v.z);
  unsafeAtomicAdd(p + 3, v.w);
}

template <>
__global__ __launch_bounds__(256) void scatter_add<2>(
    float* __restrict__ agg, const float* __restrict__ h,
    const long long* __restrict__ src, const long long* __restrict__ dst,
    int E) {
  const int lane = threadIdx.x & 31;
  const int e = __builtin_amdgcn_readfirstlane(
      blockIdx.x * (blockDim.x >> 5) + (threadIdx.x >> 5));
  if (e >= E) return;
  const int s = __builtin_amdgcn_readfirstlane((int)src[e]);
  const int d = __builtin_amdgcn_readfirstlane((int)dst[e]);
  const int f0 = lane * 2;
  const v2f v = *(const v2f*)(h + (size_t)s * 64 + f0);
  float* p = agg + (size_t)d * 64 + f0;
  unsafeAtomicAdd(p + 0, v.x);
  unsafeAtomicAdd(p + 1, v.y);
}

extern "C" void kernel_launch(void* const* d_in, const int* in_sizes, int n_in,
                              void* d_out, int out_size, void* d_ws, size_t ws_size,
                              hipStream_t stream) {
  const float*     x  = (const float*)d_in[0];
  const long long* ei = (const long long*)d_in[1];  // int64 per reference
  const float*     W1 = (const float*)d_in[2];
  const float*     b1 = (const float*)d_in[3];
  const float*     W2 = (const float*)d_in[4];
  const float*     b2 = (const float*)d_in[5];
  const float*     W3 = (const float*)d_in[6];
  const float*     b3 = (const float*)d_in[7];

  const int F    = 128;                 // hidden width
  const int Fout = in_sizes[7];         // 64 (b3 length)
  const int Nn   = in_sizes[0] / F;     // 100000 nodes
  const int E    = in_sizes[1] / 2;     // 640000 edges
  const long long* srcIdx = ei;
  const long long* dstIdx = ei + E;

  float* h   = (float*)d_ws;                  // [Nn, 128] scratch
  float* agg = h + (size_t)Nn * F;            // [Nn, 128] scratch
  float* out = (float*)d_out;                 // [Nn, 64]

  const dim3 blk(128, 1, 1);
  const dim3 gridG((Nn + 31) / 32, 1, 1);     // 32-row x 128-col block tile

  const int totH4 = Nn * F / 4;
  const int totO4 = Nn * Fout / 4;
  const dim3 ibH((totH4 + 255) / 256, 1, 1);
  const dim3 ibO((totO4 + 255) / 256, 1, 1);
  const dim3 sgrid((E + 7) / 8, 1, 1);        // 8 waves/block, 1 wave/edge

  // ----- Layer 1: h = x @ W1 ; agg = b1 + scatter(h) -----
  gemm_wmma_f32<false><<<gridG, blk, 0, stream>>>(x, W1, h, Nn, F, F);
  init_bias4<<<ibH, 256, 0, stream>>>(agg, b1, totH4, F / 4 - 1);
  scatter_add<4><<<sgrid, 256, 0, stream>>>(agg, h, srcIdx, dstIdx, E);

  // ----- Layer 2: h = relu(agg) @ W2 ; agg = b2 + scatter(h) -----
  gemm_wmma_f32<true><<<gridG, blk, 0, stream>>>(agg, W2, h, Nn, F, F);
  init_bias4<<<ibH, 256, 0, stream>>>(agg, b2, totH4, F / 4 - 1);
  scatter_add<4><<<sgrid, 256, 0, stream>>>(agg, h, srcIdx, dstIdx, E);

  // ----- Layer 3: h = relu(agg) @ W3 ; out = b3 + scatter(h) -----
  gemm_wmma_f32<true><<<gridG, blk, 0, stream>>>(agg, W3, h, Nn, Fout, F);
  init_bias4<<<ibO, 256, 0, stream>>>(out, b3, totO4, Fout / 4 - 1);
  scatter_add<2><<<sgrid, 256, 0, stream>>>(out, h, srcIdx, dstIdx, E);
}